// GAT_2963527434975
// MI455X (gfx1250) — compile-verified
//
#include <hip/hip_runtime.h>
#include <hip/hip_bf16.h>

typedef __attribute__((ext_vector_type(2))) float v2f;
typedef __attribute__((ext_vector_type(8))) float v8f;

#define NNODES 50000
#define NEDGES 800000
#define CH 256            // H*D is 256 for every layer
#define NEG_SLOPE 0.2f

// ---------- helpers: monotonic float<->uint key for atomicMax ----------
__device__ __forceinline__ unsigned fkey(float f) {
  unsigned u = __float_as_uint(f);
  return (u & 0x80000000u) ? ~u : (u | 0x80000000u);
}
__device__ __forceinline__ float funkey(unsigned k) {
  return __uint_as_float((k & 0x80000000u) ? (k & 0x7FFFFFFFu) : ~k);
}

// ---------- zero fill (float4 granular; n4 = count of float4) ----------
__global__ void fill_zero4(float4* __restrict__ p, int n4) {
  int i = blockIdx.x * blockDim.x + threadIdx.x;
  if (i < n4) p[i] = make_float4(0.f, 0.f, 0.f, 0.f);
}

// ---------- GEMM: C[N,256] = A[N,K] @ W[K,256] via V_WMMA_F32_16X16X4_F32 ----------
// block = 256 threads = 8 waves; block b owns rows [16b,16b+16); wave w owns cols [32w,32w+32)
__global__ void gemm_f32_wmma(const float* __restrict__ A,
                              const float* __restrict__ W,
                              float* __restrict__ C, int K) {
  const int lane = threadIdx.x & 31;
  const int wave = threadIdx.x >> 5;      // 0..7
  const int r16  = lane & 15;
  const int hi   = lane >> 4;             // 0 or 1 (K sub-pair)
  const size_t rowBase = (size_t)blockIdx.x * 16;

  const int col = wave * 32 + r16;
  const float* Arow = A + (rowBase + r16) * (size_t)K + hi * 2;
  const float* Wp0  = W + col;
  const float* Wp1  = W + col + 16;

  v8f acc0 = {};
  v8f acc1 = {};
  for (int k = 0; k < K; k += 4) {
    // A 16x4 fragment: lanes 0-15 K={k,k+1}, lanes 16-31 K={k+2,k+3}
    v2f a = *(const v2f*)(Arow + k);
    const int kr = k + hi * 2;
    v2f b0, b1;
    b0.x = Wp0[(size_t)kr * CH];
    b0.y = Wp0[(size_t)(kr + 1) * CH];
    b1.x = Wp1[(size_t)kr * CH];
    b1.y = Wp1[(size_t)(kr + 1) * CH];
    acc0 = __builtin_amdgcn_wmma_f32_16x16x4_f32(false, a, false, b0, (short)0, acc0, false, false);
    acc1 = __builtin_amdgcn_wmma_f32_16x16x4_f32(false, a, false, b1, (short)0, acc1, false, false);
  }
  // C/D layout: lanes 0-15 -> rows 0..7, lanes 16-31 -> rows 8..15; col = lane&15
  float* Cp = C + (rowBase + (size_t)hi * 8) * CH + col;
  #pragma unroll
  for (int i = 0; i < 8; ++i) {
    Cp[(size_t)i * CH]      = acc0[i];
    Cp[(size_t)i * CH + 16] = acc1[i];
  }
}

// ---------- per-node attention dots: el/er[n,h] = sum_d F[n,h*D+d] * al/ar[h*D+d] ----------
__global__ void node_dots(const float* __restrict__ F, const float* __restrict__ al,
                          const float* __restrict__ ar, float* __restrict__ el,
                          float* __restrict__ er, int H, int Hshift, int D) {
  int i = blockIdx.x * blockDim.x + threadIdx.x;  // n*H + h
  if (i >= NNODES * H) return;
  int n = i >> Hshift, h = i & (H - 1);
  const float4* f = (const float4*)(F + (size_t)n * CH + h * D);
  const float4* a = (const float4*)(al + h * D);
  const float4* b = (const float4*)(ar + h * D);
  float sl = 0.f, sr = 0.f;
  int D4 = D >> 2;
  for (int d = 0; d < D4; ++d) {
    float4 v = f[d], x = a[d], y = b[d];
    sl += v.x * x.x + v.y * x.y + v.z * x.z + v.w * x.w;
    sr += v.x * y.x + v.y * y.y + v.z * y.z + v.w * y.w;
  }
  el[i] = sl; er[i] = sr;
}

// ---------- edge pass 1: e = leaky(el[src]+er[dst]); store; atomicMax per dst ----------
__global__ void edge_score(const int* __restrict__ src, const int* __restrict__ dst,
                           const float* __restrict__ el, const float* __restrict__ er,
                           float* __restrict__ ex, unsigned* __restrict__ mkey,
                           int H, int Hshift) {
  int i = blockIdx.x * blockDim.x + threadIdx.x;  // e*H + h
  if (i >= NEDGES * H) return;
  int e = i >> Hshift, h = i & (H - 1);
  int s = src[e], d = dst[e];
  float v = el[(s << Hshift) + h] + er[(d << Hshift) + h];
  v = v > 0.f ? v : NEG_SLOPE * v;
  ex[i] = v;
  atomicMax(&mkey[(d << Hshift) + h], fkey(v));
}

// ---------- edge pass 2: ex = exp(e - m[dst]); atomicAdd per-dst sum ----------
__global__ void edge_exp(const int* __restrict__ dst, const unsigned* __restrict__ mkey,
                         float* __restrict__ ex, float* __restrict__ ssum,
                         int H, int Hshift) {
  int i = blockIdx.x * blockDim.x + threadIdx.x;
  if (i >= NEDGES * H) return;
  int e = i >> Hshift, h = i & (H - 1);
  int d = dst[e];
  float x = __expf(ex[i] - funkey(mkey[(d << Hshift) + h]));
  ex[i] = x;
  atomicAdd(&ssum[(d << Hshift) + h], x);
}

// ---------- edge pass 3: msg[dst,c] += (ex/s) * F[src,c]; one wave per edge ----------
// lane handles channels [4*lane, 4*lane+4) and [128+4*lane, 128+4*lane+4)
// (a 4-channel chunk never crosses a 64-channel head boundary)
__global__ void edge_agg(const int* __restrict__ src, const int* __restrict__ dst,
                         const float* __restrict__ F, const float* __restrict__ ex,
                         const float* __restrict__ ssum, float* __restrict__ out,
                         int H, int shiftD) {
  int lane = threadIdx.x & 31;
  int widx = blockIdx.x * (blockDim.x >> 5) + (threadIdx.x >> 5);  // edge index
  if (widx >= NEDGES) return;
  int s = src[widx], d = dst[widx];
  const float* fs = F + (size_t)s * CH;
  float* od = out + (size_t)d * CH;

  int c0 = lane * 4;
  int c1 = 128 + lane * 4;
  int h0 = c0 >> shiftD;
  int h1 = c1 >> shiftD;
  float a0 = ex[(size_t)widx * H + h0] / ssum[(size_t)d * H + h0];
  float a1 = ex[(size_t)widx * H + h1] / ssum[(size_t)d * H + h1];

  float4 v0 = *(const float4*)(fs + c0);
  float4 v1 = *(const float4*)(fs + c1);
  atomicAdd(&od[c0 + 0], a0 * v0.x);
  atomicAdd(&od[c0 + 1], a0 * v0.y);
  atomicAdd(&od[c0 + 2], a0 * v0.z);
  atomicAdd(&od[c0 + 3], a0 * v0.w);
  atomicAdd(&od[c1 + 0], a1 * v1.x);
  atomicAdd(&od[c1 + 1], a1 * v1.y);
  atomicAdd(&od[c1 + 2], a1 * v1.z);
  atomicAdd(&od[c1 + 3], a1 * v1.w);
}

// ---------- bias (+ optional relu), in place, float4 ----------
__global__ void bias_act(float4* __restrict__ x, const float4* __restrict__ b, int relu) {
  int i = blockIdx.x * blockDim.x + threadIdx.x;   // float4 index
  if (i >= NNODES * (CH / 4)) return;
  float4 v = x[i];
  float4 bb = b[i & (CH / 4 - 1)];
  v.x += bb.x; v.y += bb.y; v.z += bb.z; v.w += bb.w;
  if (relu) {
    v.x = v.x < 0.f ? 0.f : v.x;
    v.y = v.y < 0.f ? 0.f : v.y;
    v.z = v.z < 0.f ? 0.f : v.z;
    v.w = v.w < 0.f ? 0.f : v.w;
  }
  x[i] = v;
}

// ---------- one GAT layer ----------
static void run_layer(const float* Hin, int K, const float* Wt, const float* al,
                      const float* ar, const float* b, int H, int Hshift, int D, int shiftD,
                      float* F, float* Mout, float* el, float* er, unsigned* mkey,
                      float* ssum, float* ex, const int* src, const int* dst,
                      int relu, hipStream_t stream) {
  gemm_f32_wmma<<<NNODES / 16, 256, 0, stream>>>(Hin, Wt, F, K);
  int nh = NNODES * H;
  node_dots<<<(nh + 255) / 256, 256, 0, stream>>>(F, al, ar, el, er, H, Hshift, D);
  // zero mkey + ssum (contiguous, N*4 each) and the message accumulator
  fill_zero4<<<(NNODES * 2 + 255) / 256, 256, 0, stream>>>((float4*)mkey, NNODES * 2);
  fill_zero4<<<(NNODES * (CH / 4) + 255) / 256, 256, 0, stream>>>((float4*)Mout, NNODES * (CH / 4));
  int eh = NEDGES * H;
  edge_score<<<(eh + 255) / 256, 256, 0, stream>>>(src, dst, el, er, ex, mkey, H, Hshift);
  edge_exp<<<(eh + 255) / 256, 256, 0, stream>>>(dst, mkey, ex, ssum, H, Hshift);
  edge_agg<<<NEDGES / 8, 256, 0, stream>>>(src, dst, F, ex, ssum, Mout, H, shiftD);
  bias_act<<<(NNODES * (CH / 4) + 255) / 256, 256, 0, stream>>>((float4*)Mout, (const float4*)b, relu);
}

extern "C" void kernel_launch(void* const* d_in, const int* in_sizes, int n_in,
                              void* d_out, int out_size, void* d_ws, size_t ws_size,
                              hipStream_t stream) {
  const float* feats = (const float*)d_in[0];
  const int*   src   = (const int*)d_in[1];
  const int*   dst   = (const int*)d_in[2];
  const float* W0 = (const float*)d_in[3];
  const float* al0 = (const float*)d_in[4];
  const float* ar0 = (const float*)d_in[5];
  const float* b0 = (const float*)d_in[6];
  const float* W1 = (const float*)d_in[7];
  const float* al1 = (const float*)d_in[8];
  const float* ar1 = (const float*)d_in[9];
  const float* b1 = (const float*)d_in[10];
  const float* W2 = (const float*)d_in[11];
  const float* al2 = (const float*)d_in[12];
  const float* ar2 = (const float*)d_in[13];
  const float* b2 = (const float*)d_in[14];
  float* out = (float*)d_out;

  float* ws = (float*)d_ws;
  float* F  = ws;                                  // [N,256] projected feats
  float* M  = F + (size_t)NNODES * CH;             // [N,256] msg / hidden
  float* el = M + (size_t)NNODES * CH;             // [N,4]
  float* er = el + (size_t)NNODES * 4;             // [N,4]
  unsigned* mkey = (unsigned*)(er + (size_t)NNODES * 4);  // [N,4]
  float* ssum = (float*)mkey + (size_t)NNODES * 4;        // [N,4]
  float* ex   = ssum + (size_t)NNODES * 4;                // [E,4]

  // layer 0: 512 -> 4x64, relu
  run_layer(feats, 512, W0, al0, ar0, b0, 4, 2, 64, 6,
            F, M, el, er, mkey, ssum, ex, src, dst, 1, stream);
  // layer 1: 256 -> 4x64, relu
  run_layer(M, 256, W1, al1, ar1, b1, 4, 2, 64, 6,
            F, M, el, er, mkey, ssum, ex, src, dst, 1, stream);
  // layer 2: 256 -> 1x256, no act; mean over single head == identity; write d_out
  run_layer(M, 256, W2, al2, ar2, b2, 1, 0, 256, 8,
            F, out, el, er, mkey, ssum, ex, src, dst, 0, stream);
}